// SymbolicTraversal_8443905704284
// MI455X (gfx1250) — compile-verified
//
#include <hip/hip_runtime.h>
#include <stdint.h>

#define NREL_MAX 32   // LDS mask table size (N_REL=16 in reference; pad for safety)

// ---------------------------------------------------------------------------
// CDNA5 native fp32 atomic max (GLOBAL_ATOMIC_MAX_NUM_F32, no-return form).
// Device scope: output rows are shared across all workgroups.
// IEEE maximumNumber semantics; inputs here are non-NaN, non-negative.
// ---------------------------------------------------------------------------
__device__ __forceinline__ void atomic_fmax_f32(float* addr, float v) {
    unsigned long long a = (unsigned long long)(uintptr_t)addr;
    asm volatile("global_atomic_max_num_f32 %0, %1, off scope:SCOPE_DEV"
                 :
                 : "v"(a), "v"(v)
                 : "memory");
}

// ---------------------------------------------------------------------------
// Zero-fill d_out (reference clamps at 0, and all messages are >= 0, so 0 is
// the correct identity). Must run every call: harness poisons once and never
// re-poisons between graph replays.
// ---------------------------------------------------------------------------
__global__ void st_zero_out(float* __restrict__ out, int n) {
    int i = 4 * (blockIdx.x * blockDim.x + threadIdx.x);
    if (i + 3 < n) {
        *(float4*)(out + i) = make_float4(0.f, 0.f, 0.f, 0.f);
    } else {
        for (; i < n; ++i) out[i] = 0.f;
    }
}

// ---------------------------------------------------------------------------
// One pass over edges, 4 edges per thread via B128 loads.
// Per block: build relation -> batch bitmask table in LDS (ds_or_b32),
// then for each edge expand to matching batches (avg 2 of 32) and
// gather h_prob (L2-resident) + fp32 atomic-max scatter to out (L2-resident).
// ---------------------------------------------------------------------------
__device__ __forceinline__ void process_edge(
        int head, int tail, int rel, float w,
        const unsigned* __restrict__ relmask,
        const float* __restrict__ h_prob,
        float* __restrict__ out, int N) {
    unsigned mask = relmask[rel & (NREL_MAX - 1)];
    while (mask) {
        int b = __builtin_ctz(mask);
        mask &= mask - 1;
        float h = h_prob[b * N + head];   // 6.4 MB table -> L2 hit
        float m = w * h;                  // single fp32 mul, RTE (matches JAX)
        atomic_fmax_f32(&out[b * N + tail], m);
    }
}

__global__ void st_edge_scatter_max(
        const float* __restrict__ h_prob,
        const float* __restrict__ edge_weight,
        const int*   __restrict__ heads,
        const int*   __restrict__ tails,
        const int*   __restrict__ rels,
        const int*   __restrict__ r_index,
        float*       __restrict__ out,
        int E, int N, int B) {
    __shared__ unsigned relmask[NREL_MAX];

    // Build relation -> batch-bitmask table (B <= 32 lanes of mask).
    if (threadIdx.x < NREL_MAX) relmask[threadIdx.x] = 0u;
    __syncthreads();
    if ((int)threadIdx.x < B) {
        int r = r_index[threadIdx.x];
        atomicOr(&relmask[r & (NREL_MAX - 1)], 1u << threadIdx.x);  // ds_or_b32
    }
    __syncthreads();

    int base = 4 * (blockIdx.x * blockDim.x + threadIdx.x);
    if (base >= E) return;

    if (base + 3 < E) {
        // 4 edges per thread: four global_load_b128 streams (fully coalesced)
        int4   h4 = *(const int4*)(heads + base);
        int4   t4 = *(const int4*)(tails + base);
        int4   r4 = *(const int4*)(rels + base);
        float4 w4 = *(const float4*)(edge_weight + base);

        int   hs[4] = {h4.x, h4.y, h4.z, h4.w};
        int   ts[4] = {t4.x, t4.y, t4.z, t4.w};
        int   rs[4] = {r4.x, r4.y, r4.z, r4.w};
        float ws[4] = {w4.x, w4.y, w4.z, w4.w};

#pragma unroll
        for (int j = 0; j < 4; ++j)
            process_edge(hs[j], ts[j], rs[j], ws[j], relmask, h_prob, out, N);
    } else {
        for (int e = base; e < E; ++e)
            process_edge(heads[e], tails[e], rels[e], edge_weight[e],
                         relmask, h_prob, out, N);
    }
}

// ---------------------------------------------------------------------------
// Launch. Inputs (setup_inputs order):
//   d_in[0] h_prob      float32 [B*N]
//   d_in[1] edge_weight float32 [E]
//   d_in[2] heads       int32   [E]
//   d_in[3] tails       int32   [E]
//   d_in[4] rels        int32   [E]
//   d_in[5] r_index     int32   [B]
// d_out: float32 [B*N]
// ---------------------------------------------------------------------------
extern "C" void kernel_launch(void* const* d_in, const int* in_sizes, int n_in,
                              void* d_out, int out_size, void* d_ws, size_t ws_size,
                              hipStream_t stream) {
    const float* h_prob      = (const float*)d_in[0];
    const float* edge_weight = (const float*)d_in[1];
    const int*   heads       = (const int*)d_in[2];
    const int*   tails       = (const int*)d_in[3];
    const int*   rels        = (const int*)d_in[4];
    const int*   r_index     = (const int*)d_in[5];
    float*       out         = (float*)d_out;

    const int B = in_sizes[5];
    const int E = in_sizes[1];
    const int N = in_sizes[0] / B;

    const int threads = 256;  // 8 wave32 waves per block

    // Pass 1: zero the output (identity of clamped max-semiring).
    int zwork   = (out_size + 3) / 4;
    int zblocks = (zwork + threads - 1) / threads;
    st_zero_out<<<zblocks, threads, 0, stream>>>(out, out_size);

    // Pass 2: single sweep over all edges, 4 edges/thread.
    int ework   = (E + 3) / 4;
    int eblocks = (ework + threads - 1) / threads;
    st_edge_scatter_max<<<eblocks, threads, 0, stream>>>(
        h_prob, edge_weight, heads, tails, rels, r_index, out, E, N, B);
}